// S4Block_5746666242102
// MI455X (gfx1250) — compile-verified
//
#include <hip/hip_runtime.h>
#include <math.h>

// ---------------- problem constants ----------------
namespace {
constexpr int kCH  = 256;          // channels (NSSM == CH, REP == 1)
constexpr int kNH  = 32;           // half state dim
constexpr int kNH2 = 64;           // 2*NH
constexpr int kL   = 16384;        // seq len
constexpr int kNF  = kL / 2 + 1;   // 8193 rfft bins
constexpr int kN   = kL / 2;       // 8192-point complex inverse FFT

// tiny workspace: only the per-channel (A*dt) poles and v = Bc(x)Cc*dt tables
// cross between the two fused kernels. Everything else stays in LDS.
constexpr size_t OFF_ADT = 0;                              // kCH*kNH cplx
constexpr size_t OFF_V   = (size_t)kCH * kNH * 2;          // 4*kCH*kNH cplx
}

// ---------------- complex helpers (float2) ----------------
__device__ __forceinline__ float2 cmulf(float2 a, float2 b) {
  return make_float2(a.x * b.x - a.y * b.y, a.x * b.y + a.y * b.x);
}
__device__ __forceinline__ float2 caddf(float2 a, float2 b) { return make_float2(a.x + b.x, a.y + b.y); }
__device__ __forceinline__ float2 csubf(float2 a, float2 b) { return make_float2(a.x - b.x, a.y - b.y); }
__device__ __forceinline__ float2 conjf2(float2 a) { return make_float2(a.x, -a.y); }
__device__ __forceinline__ float2 cinvf(float2 a) {
  float id = 1.0f / (a.x * a.x + a.y * a.y);
  return make_float2(a.x * id, -a.y * id);
}
__device__ __forceinline__ float2 cscalef(float2 a, float s) { return make_float2(a.x * s, a.y * s); }
__device__ __forceinline__ float2 ld2(const float* p, size_t i) { return make_float2(p[2 * i], p[2 * i + 1]); }
__device__ __forceinline__ void   st2(float* p, size_t i, float2 v) { p[2 * i] = v.x; p[2 * i + 1] = v.y; }

// ---------------- WMMA types (CDNA5, wave32) ----------------
typedef float v2f __attribute__((ext_vector_type(2)));
typedef float v8f __attribute__((ext_vector_type(8)));

// ============================================================
// Kernel 1 (fused): per channel h (one block, 8 waves):
//   A) build A_bar[64][64] (complex) directly in LDS
//      A_bar[m][n] = D2[m]*( E2[m]*(m==n) - P2[m]*(Q2[n] + S[n]) )
//      S[n] = R2[n]E2[n] - Q2[n]*sum_m(R2[m]P2[m])
//   B) dA_L = A_bar^(2^14): 14 LDS-resident complex squarings via
//      V_WMMA_F32_16X16X4_F32 (Re=RR-II, Im=RI+IR), ping-pong in LDS
//   C) C_tilde = (C_t - dA_L^T C_t)[:NH]; v = Bc(x)Cc*dt -> ws
// dynamic LDS: 2 buffers x (Re+Im) x 64x64 f32 = 64 KB
// ============================================================
__global__ __launch_bounds__(256) void s4_matpow_kernel(
    const float* __restrict__ A_real, const float* __restrict__ A_imag,
    const float* __restrict__ P_ri,  const float* __restrict__ B_ri,
    const float* __restrict__ C_ri,  const float* __restrict__ dt_raw,
    float* __restrict__ ws) {
  extern __shared__ float smem[];            // 4 * 4096 floats
  // plane layout: [0)=buf0 Re, [4096)=buf0 Im, [8192)=buf1 Re, [12288)=buf1 Im

  __shared__ float2 sD[kNH2], sE[kNH2], sP[kNH2], sQ[kNH2], sR[kNH2], sS[kNH2];
  __shared__ float2 sCt[kNH2], sB[kNH], rp[kNH];
  __shared__ float  red[kNH];
  __shared__ float  sRm, sDt;
  __shared__ float2 sSrp;

  const int h = blockIdx.x;
  const int t = threadIdx.x;

  // ---------------- Phase A: prep ----------------
  const float xr  = dt_raw[h];
  const float dtv = (xr > 20.0f) ? xr : log1pf(expf(xr));   // softplus
  const float i2dt = 2.0f / dtv;
  if (t == 0) sDt = dtv;

  if (t < kNH) {
    const size_t gi = (size_t)h * kNH + t;
    const float ar = A_real[gi], ai = A_imag[gi];
    const float ea = expf(ar);
    const float2 A = make_float2(-ea, -ai);                  // -exp(Ar) - i*Ai
    const float2 P = ld2(P_ri, gi);
    const float2 B = ld2(B_ri, gi);
    const float2 C = ld2(C_ri, gi);
    const float2 Q = conjf2(P);
    const float2 D = cinvf(make_float2(i2dt - A.x, -A.y));   // 1/(2/dt - A)
    const float2 E = make_float2(i2dt + A.x, A.y);           // 2/dt + A
    sD[t] = D; sE[t] = E; sP[t] = P; sQ[t] = Q; sB[t] = B;
    sCt[t]       = C;
    sCt[t + kNH] = conjf2(C);
    red[t] = cmulf(cmulf(Q, D), P).x;                        // Re(Q*D*P)
    st2(ws + OFF_ADT, gi, make_float2(A.x * dtv, A.y * dtv));
  }
  __syncthreads();

  if (t == 0) {
    float s = 0.0f;
    for (int n = 0; n < kNH; ++n) s += red[n];
    sRm = 1.0f + 2.0f * s;
  }
  __syncthreads();

  if (t < kNH) {
    const float irm = 1.0f / sRm;
    const float2 R = cscalef(cmulf(sQ[t], sD[t]), irm);
    sR[t] = R;
    rp[t] = cmulf(R, sP[t]);
    sD[t + kNH] = conjf2(sD[t]);
    sE[t + kNH] = conjf2(sE[t]);
    sP[t + kNH] = conjf2(sP[t]);
    sQ[t + kNH] = conjf2(sQ[t]);
    sR[t + kNH] = conjf2(R);
  }
  __syncthreads();

  if (t == 0) {
    float2 s = make_float2(0.0f, 0.0f);
    for (int n = 0; n < kNH; ++n) s.x += 2.0f * rp[n].x;     // rp + conj(rp)
    sSrp = s;
  }
  __syncthreads();

  if (t < kNH2) sS[t] = csubf(cmulf(sR[t], sE[t]), cmulf(sQ[t], sSrp));
  __syncthreads();

  if (t < kNH2) {                                            // row m = t of A_bar
    const float2 Dm = sD[t], Em = sE[t], Pm = sP[t];
    for (int n = 0; n < kNH2; ++n) {
      float2 v = cmulf(Pm, caddf(sQ[n], sS[n]));
      v = make_float2(-v.x, -v.y);
      if (n == t) { v.x += Em.x; v.y += Em.y; }
      v = cmulf(Dm, v);
      smem[t * kNH2 + n]        = v.x;                       // buf0 Re
      smem[4096 + t * kNH2 + n] = v.y;                       // buf0 Im
    }
  }
  __syncthreads();

  // ---------------- Phase B: 14 LDS-resident complex squarings ----------------
  const int lane = t & 31;
  const int wave = t >> 5;
  const int lm = lane & 15;      // M/N within 16x16 tile
  const int lh = lane >> 4;      // half-wave (K grouping)
  int cur = 0;
  for (int it = 0; it < 14; ++it) {
    const float* sRe = (cur == 0) ? smem            : smem + 8192;
    const float* sIm = (cur == 0) ? smem + 4096     : smem + 12288;
    float*       dRe = (cur == 0) ? smem + 8192     : smem;
    float*       dIm = (cur == 0) ? smem + 12288    : smem + 4096;
    // 16 output tiles over 8 waves -> exactly 2 tiles per wave; constant trip
    // count keeps control flow scalar (no exec masking around WMMA).
#pragma unroll
    for (int rep = 0; rep < 2; ++rep) {
      const int tile = wave + rep * 8;
      const int ti = tile >> 2, tj = tile & 3;
      v8f accRR = {0.f,0.f,0.f,0.f,0.f,0.f,0.f,0.f};
      v8f accII = accRR, accRI = accRR, accIR = accRR;
      const int am = ti * 16 + lm;                           // A row
      const int bn = tj * 16 + lm;                           // B col
      for (int k = 0; k < kNH2; k += 4) {
        const int ak = k + lh * 2;     // A frag: lanes 0-15 K,K+1; 16-31 K+2,K+3
        v2f aRe = { sRe[am * kNH2 + ak], sRe[am * kNH2 + ak + 1] };
        v2f aIm = { sIm[am * kNH2 + ak], sIm[am * kNH2 + ak + 1] };
        const int bk = k + lh;         // B frag: rows striped across half-waves
        v2f bRe_ = { sRe[bk * kNH2 + bn], sRe[(bk + 2) * kNH2 + bn] };
        v2f bIm_ = { sIm[bk * kNH2 + bn], sIm[(bk + 2) * kNH2 + bn] };
        accRR = __builtin_amdgcn_wmma_f32_16x16x4_f32(false, aRe, false, bRe_, (short)0, accRR, false, false);
        accII = __builtin_amdgcn_wmma_f32_16x16x4_f32(false, aIm, false, bIm_, (short)0, accII, false, false);
        accRI = __builtin_amdgcn_wmma_f32_16x16x4_f32(false, aRe, false, bIm_, (short)0, accRI, false, false);
        accIR = __builtin_amdgcn_wmma_f32_16x16x4_f32(false, aIm, false, bRe_, (short)0, accIR, false, false);
      }
      const v8f oRe = accRR - accII;
      const v8f oIm = accRI + accIR;
      // C/D layout: VGPR r -> M = 16*ti + r + 8*(lane>=16), N = 16*tj + (lane&15)
      const int mBase = ti * 16 + lh * 8;
#pragma unroll
      for (int r = 0; r < 8; ++r) {
        const int m = mBase + r;
        dRe[m * kNH2 + bn] = oRe[r];
        dIm[m * kNH2 + bn] = oIm[r];
      }
    }
    __syncthreads();
    cur ^= 1;
  }

  // ---------------- Phase C: C_tilde + v ----------------
  if (t < kNH) {
    const float* fRe = (cur == 0) ? smem        : smem + 8192;
    const float* fIm = (cur == 0) ? smem + 4096 : smem + 12288;
    float2 acc = make_float2(0.0f, 0.0f);
    for (int n = 0; n < kNH2; ++n) {                          // prod[m]=sum dA_L[n][m]*C_t[n]
      const float2 a = make_float2(fRe[n * kNH2 + t], fIm[n * kNH2 + t]);
      acc = caddf(acc, cmulf(a, sCt[n]));
    }
    const float2 ctil = csubf(sCt[t], acc);
    const float2 P = sP[t], Q = sQ[t], B = sB[t];
    const float d = sDt;
    const size_t gi = (size_t)h * kNH + t;
    const size_t vs = (size_t)kCH * kNH;
    st2(ws + OFF_V, 0 * vs + gi, cscalef(cmulf(B, ctil), d));   // v00
    st2(ws + OFF_V, 1 * vs + gi, cscalef(cmulf(B, Q),    d));   // v01
    st2(ws + OFF_V, 2 * vs + gi, cscalef(cmulf(P, ctil), d));   // v10
    st2(ws + OFF_V, 3 * vs + gi, cscalef(cmulf(P, Q),    d));   // v11
  }
}

// ============================================================
// Kernel 2 (fused): per channel h (one block, 8 waves):
//   D) conjugate-symmetric Cauchy + rank-1 Woodbury -> K[f] in LDS (8193 bins)
//   E) Hermitian pack: Xe[k]=(H[k]+conj(H[N-k]))/2,
//      Xo[k]=e^{+i*pi*k/N}(H[k]-conj(H[N-k]))/2, Z=Xe+i*Xo
//      (imag of DC/Nyquist dropped, matching c2r semantics)
//   F) N=8192 inverse radix-2 Stockham FFT, ping-pong Z-buf <-> K-buf
//   G) x[2t]=Re(y)/N, x[2t+1]=Im(y)/N
// dynamic LDS: (8192 + 8193) cplx f32 = 131080 B -> 2 WGs/WGP
// ============================================================
__global__ __launch_bounds__(256) void s4_spectrum_kernel(
    const float* __restrict__ ws, float* __restrict__ out) {
  extern __shared__ float smem2[];
  float2* bufZ = (float2*)smem2;       // 8192 entries
  float2* bufK = bufZ + kN;            // 8193 entries (K, then FFT ping-pong)
  __shared__ float2 sAdt[kNH];
  __shared__ float2 sV[4 * kNH];

  const int h = blockIdx.x, t = threadIdx.x;

  // stage per-channel pole/residue tables once
  if (t < kNH) sAdt[t] = ld2(ws + OFF_ADT, (size_t)h * kNH + t);
  if (t < 4 * kNH) {
    const int ij = t >> 5, n = t & 31;
    sV[t] = ld2(ws + OFF_V, (size_t)ij * kCH * kNH + (size_t)h * kNH + n);
  }
  __syncthreads();

  // ---------------- Phase D: Cauchy + Woodbury ----------------
  for (int f = t; f < kNF; f += 256) {
    const float th = -2.0f * (float)M_PI * (float)f / (float)kL;
    float so, co;
    __sincosf(th, &so, &co);
    const float2 onep = make_float2(1.0f + co, so);          // 1 + omega
    const float2 onem = make_float2(1.0f - co, -so);         // 1 - omega
    const float2 ip   = cinvf(onep);
    const float2 z    = cmulf(make_float2(2.0f * onem.x, 2.0f * onem.y), ip);

    float2 r00 = make_float2(0.f, 0.f), r01 = r00, r10 = r00, r11 = r00;
#pragma unroll 4
    for (int n = 0; n < kNH; ++n) {
      const float2 w  = sAdt[n];
      const float2 c1 = cinvf(make_float2(z.x - w.x, z.y - w.y));   // 1/(z-w)
      const float2 c2 = cinvf(make_float2(z.x - w.x, z.y + w.y));   // 1/(z-conj w)
      const float2 v00 = sV[0 * kNH + n];
      const float2 v01 = sV[1 * kNH + n];
      const float2 v10 = sV[2 * kNH + n];
      const float2 v11 = sV[3 * kNH + n];
      r00 = caddf(r00, caddf(cmulf(v00, c1), cmulf(conjf2(v00), c2)));
      r01 = caddf(r01, caddf(cmulf(v01, c1), cmulf(conjf2(v01), c2)));
      r10 = caddf(r10, caddf(cmulf(v10, c1), cmulf(conjf2(v10), c2)));
      r11 = caddf(r11, caddf(cmulf(v11, c1), cmulf(conjf2(v11), c2)));
    }
    const float2 den = make_float2(1.0f + r11.x, r11.y);
    float2 K = csubf(r00, cmulf(cmulf(r01, r10), cinvf(den)));
    K = cmulf(K, make_float2(2.0f * ip.x, 2.0f * ip.y));            // * 2/(1+omega)
    bufK[f] = K;
  }
  __syncthreads();

  // ---------------- Phase E: Hermitian pack into Z ----------------
  for (int k = t; k < kN; k += 256) {
    const float2 Hk = bufK[k];
    const float2 Hj = bufK[kN - k];                 // k==0 -> Nyquist bin 8192
    const float hi = (k == 0) ? 0.0f : Hk.y;
    const float gi = (k == 0) ? 0.0f : Hj.y;
    const float xer = 0.5f * (Hk.x + Hj.x), xei = 0.5f * (hi - gi);
    const float xor_ = 0.5f * (Hk.x - Hj.x), xoi = 0.5f * (hi + gi);
    float s, c;
    __sincosf((float)M_PI * (float)k / (float)kN, &s, &c);   // e^{+2pi i k / L}
    const float Xor = xor_ * c - xoi * s;
    const float Xoi = xor_ * s + xoi * c;
    bufZ[k] = make_float2(xer - Xoi, xei + Xor);             // Xe + i*Xo
  }
  __syncthreads();

  // ---------------- Phase F: inverse Stockham FFT (13 radix-2 stages) ----------------
  int n = kN, s = 1, cur = 0;
  while (n > 1) {
    const int m = n >> 1;
    const float2* src = (cur == 0) ? bufZ : bufK;
    float2*       dst = (cur == 0) ? bufK : bufZ;
    const float theta = 2.0f * (float)M_PI / (float)n;       // + sign: inverse
    for (int b = t; b < kN / 2; b += 256) {
      const int p = b / s;
      const int q = b - p * s;
      const float2 a  = src[q + s * p];
      const float2 c2 = src[q + s * (p + m)];
      const float2 sum = make_float2(a.x + c2.x, a.y + c2.y);
      const float2 dif = make_float2(a.x - c2.x, a.y - c2.y);
      float ss, cc;
      __sincosf(theta * (float)p, &ss, &cc);
      dst[q + s * (2 * p)]     = sum;
      dst[q + s * (2 * p + 1)] = make_float2(dif.x * cc - dif.y * ss,
                                             dif.x * ss + dif.y * cc);
    }
    __syncthreads();
    n = m; s <<= 1; cur ^= 1;
  }

  // ---------------- Phase G: de-interleave + scale ----------------
  const float invN = 1.0f / (float)kN;
  float* o = out + (size_t)h * kL;
  const float2* res = (cur == 0) ? bufZ : bufK;
  for (int i = t; i < kN; i += 256) {
    const float2 y = res[i];
    o[2 * i]     = y.x * invN;
    o[2 * i + 1] = y.y * invN;
  }
}

// ============================================================
extern "C" void kernel_launch(void* const* d_in, const int* in_sizes, int n_in,
                              void* d_out, int out_size, void* d_ws, size_t ws_size,
                              hipStream_t stream) {
  const float* A_real = (const float*)d_in[0];
  const float* A_imag = (const float*)d_in[1];
  const float* P_ri   = (const float*)d_in[2];
  const float* B_ri   = (const float*)d_in[3];
  const float* C_ri   = (const float*)d_in[4];
  const float* dt_raw = (const float*)d_in[5];
  float* ws  = (float*)d_ws;
  float* out = (float*)d_out;

  // Kernel 1: prep + 14 LDS-resident WMMA matrix squarings + C_tilde/v
  s4_matpow_kernel<<<kCH, 256, 4 * kNH2 * kNH2 * (int)sizeof(float), stream>>>(
      A_real, A_imag, P_ri, B_ri, C_ri, dt_raw, ws);

  // Kernel 2: Cauchy + Woodbury + Hermitian pack + inverse Stockham FFT
  s4_spectrum_kernel<<<kCH, 256, (kN + kNF) * (int)sizeof(float2), stream>>>(ws, out);
}